// Classifier_31610959299310
// MI455X (gfx1250) — compile-verified
//
#include <hip/hip_runtime.h>
#include <hip/hip_bf16.h>

#define N_NODES 50000
#define N_EDGES 800000
#define N_GRAPHS 64
#define F 128

typedef __attribute__((ext_vector_type(2))) float v2f;
typedef __attribute__((ext_vector_type(8))) float v8f;

// ---------------- degree / normalization ----------------

__global__ void init_deg_k(float* deg) {
    int i = blockIdx.x * blockDim.x + threadIdx.x;
    if (i < N_NODES) deg[i] = 1.0f;  // self loop contributes 1
}

__global__ void accum_deg_k(const int* __restrict__ dst, float* __restrict__ deg) {
    int e = blockIdx.x * blockDim.x + threadIdx.x;
    if (e < N_EDGES) atomicAdd(&deg[dst[e]], 1.0f);
}

__global__ void deg_to_dinv_k(float* deg) {
    int i = blockIdx.x * blockDim.x + threadIdx.x;
    if (i < N_NODES) deg[i] = rsqrtf(deg[i]);  // deg >= 1 always (self loop)
}

// ---------------- WMMA GEMM: HS = (X @ W) * dinv[row]; ACC seeded with HS ----------------
// One block: 16 rows x 128 cols. 8 waves, wave w -> 16-col tile. K loop in steps of 4
// using V_WMMA_F32_16X16X4_F32 (fp32 dense WMMA, matches reference precision).

__launch_bounds__(256)
__global__ void gemm_scale_k(const float* __restrict__ X, const float* __restrict__ W,
                             const float* __restrict__ dinv,
                             float* __restrict__ HS, float* __restrict__ ACC) {
    __shared__ float As[16 * F];   // 8 KB A tile
    __shared__ float Ws[F * F];    // 64 KB full weight matrix

    const int row0 = blockIdx.x * 16;
    const int t = threadIdx.x;

    // Stage A tile and W into LDS (float4 vectorized, fully coalesced).
    {
        const float4* srcA = (const float4*)(X + (size_t)row0 * F);
        float4* dA = (float4*)As;
        for (int i = t; i < 16 * F / 4; i += 256) dA[i] = srcA[i];
        const float4* srcW = (const float4*)W;
        float4* dW = (float4*)Ws;
        for (int i = t; i < F * F / 4; i += 256) dW[i] = srcW[i];
    }
    __syncthreads();

    const int wave  = t >> 5;
    const int lane  = t & 31;
    const int col0  = wave * 16;
    const int ln    = lane & 15;          // M index for A, N index for B/C
    const int khalf = (lane >> 4) * 2;    // lanes 16-31 hold K+2 half

    v8f c = {};
    #pragma unroll 4
    for (int k = 0; k < F; k += 4) {
        // A 16x4 fragment: lane -> row ln, (K = k+khalf, k+khalf+1)
        v2f a, b;
        const float* ap = &As[ln * F + k + khalf];
        a.x = ap[0];
        a.y = ap[1];
        // B 4x16 fragment: lane -> col (col0+ln), rows (k+khalf, k+khalf+1)
        b.x = Ws[(k + khalf) * F + col0 + ln];
        b.y = Ws[(k + khalf + 1) * F + col0 + ln];
        c = __builtin_amdgcn_wmma_f32_16x16x4_f32(
                /*neg_a=*/false, a, /*neg_b=*/false, b,
                /*c_mod=*/(short)0, c, /*reuse_a=*/false, /*reuse_b=*/false);
    }

    // C/D layout: VGPR v -> M = v (lanes 0-15) / v+8 (lanes 16-31); N = ln.
    const int rbase = row0 + (lane >> 4) * 8;
    #pragma unroll
    for (int v = 0; v < 8; ++v) {
        const int r = rbase + v;
        const float s = dinv[r];
        const float hv = c[v] * s;
        HS[(size_t)r * F + col0 + ln]  = hv;  // scaled messages for gather
        ACC[(size_t)r * F + col0 + ln] = hv;  // self-loop term seeds accumulator
    }
}

// ---------------- edge scatter: ACC[dst] += HS[src] (one wave per edge) ----------------

__launch_bounds__(256)
__global__ void edge_scatter_k(const int* __restrict__ src, const int* __restrict__ dst,
                               const float* __restrict__ HS, float* __restrict__ ACC) {
    const int w = (blockIdx.x * blockDim.x + threadIdx.x) >> 5;  // edge id
    const int lane = threadIdx.x & 31;
    if (w >= N_EDGES) return;
    const int s = src[w];
    const int d = dst[w];
    // wave covers the full 128-float row: lane loads float4 (coalesced b128)
    const float4 hv = ((const float4*)(HS + (size_t)s * F))[lane];
    float* ap = ACC + (size_t)d * F + lane * 4;
    atomicAdd(ap + 0, hv.x);   // global_atomic_add_f32, L2-resident target
    atomicAdd(ap + 1, hv.y);
    atomicAdd(ap + 2, hv.z);
    atomicAdd(ap + 3, hv.w);
}

// ---------------- finalize: H = relu(ACC * dinv[node] + b) ----------------

__global__ void finalize_k(const float* __restrict__ ACC, const float* __restrict__ dinv,
                           const float* __restrict__ b, float* __restrict__ H) {
    int i = blockIdx.x * blockDim.x + threadIdx.x;
    if (i < N_NODES * F) {
        const int node = i >> 7;
        const int f = i & (F - 1);
        const float v = fmaf(ACC[i], dinv[node], b[f]);
        H[i] = fmaxf(v, 0.0f);
    }
}

// ---------------- pooling + classifier ----------------

__global__ void zero_pool_k(float* pooled, float* counts) {
    int i = blockIdx.x * blockDim.x + threadIdx.x;
    if (i < N_GRAPHS * F) pooled[i] = 0.0f;
    if (i < N_GRAPHS) counts[i] = 0.0f;
}

__global__ void count_nodes_k(const int* __restrict__ batch, float* __restrict__ counts) {
    int i = blockIdx.x * blockDim.x + threadIdx.x;
    if (i < N_NODES) atomicAdd(&counts[batch[i]], 1.0f);
}

__global__ void pool_sum_k(const int* __restrict__ batch, const float* __restrict__ H,
                           float* __restrict__ pooled) {
    int i = blockIdx.x * blockDim.x + threadIdx.x;
    if (i < N_NODES * F) {
        const int node = i >> 7;
        const int f = i & (F - 1);
        atomicAdd(&pooled[batch[node] * F + f], H[i]);
    }
}

__global__ void classify_k(const float* __restrict__ pooled, const float* __restrict__ counts,
                           const float* __restrict__ Wc, const float* __restrict__ bc,
                           float* __restrict__ out) {
    const int t = threadIdx.x;           // 128 threads: (graph, class)
    const int g = t >> 1;
    const int cc = t & 1;
    const float inv = 1.0f / fmaxf(counts[g], 1.0f);
    float acc = bc[cc];
    for (int f = 0; f < F; ++f)
        acc = fmaf(pooled[g * F + f] * inv, Wc[f * 2 + cc], acc);
    out[g * 2 + cc] = acc;
}

// ---------------- launcher ----------------

extern "C" void kernel_launch(void* const* d_in, const int* in_sizes, int n_in,
                              void* d_out, int out_size, void* d_ws, size_t ws_size,
                              hipStream_t stream) {
    (void)in_sizes; (void)n_in; (void)out_size; (void)ws_size;

    const float* x   = (const float*)d_in[0];
    const int*   ei  = (const int*)d_in[1];       // [2, 800000] row-major
    const int*   bat = (const int*)d_in[2];
    const float* W1  = (const float*)d_in[3];
    const float* b1  = (const float*)d_in[4];
    const float* W2  = (const float*)d_in[5];
    const float* b2  = (const float*)d_in[6];
    const float* Wc  = (const float*)d_in[7];
    const float* bc  = (const float*)d_in[8];
    float* out = (float*)d_out;

    const int* srcp = ei;
    const int* dstp = ei + N_EDGES;

    // workspace layout (256B aligned)
    char* ws = (char*)d_ws;
    float* dinv   = (float*)(ws);                                   // 50000 f
    float* HS     = (float*)(ws + 200192);                          // 6.4M f
    float* ACC    = (float*)(ws + 200192 + 25600000);               // 6.4M f
    float* H      = (float*)(ws + 200192 + 2ull * 25600000);        // 6.4M f
    float* pooled = (float*)(ws + 200192 + 3ull * 25600000);        // 8192 f
    float* counts = pooled + N_GRAPHS * F;                          // 64 f

    const int gN   = (N_NODES + 255) / 256;         // node-grid
    const int gE   = (N_EDGES + 255) / 256;         // edge-grid (thread per edge)
    const int gEW  = (N_EDGES * 32 + 255) / 256;    // wave per edge
    const int gNF  = (N_NODES * F + 255) / 256;     // element grid
    const int gRow = N_NODES / 16;                  // 3125, exact

    // symmetric degree normalization (deg includes self loop)
    init_deg_k<<<gN, 256, 0, stream>>>(dinv);
    accum_deg_k<<<gE, 256, 0, stream>>>(dstp, dinv);
    deg_to_dinv_k<<<gN, 256, 0, stream>>>(dinv);

    // layer 1
    gemm_scale_k<<<gRow, 256, 0, stream>>>(x, W1, dinv, HS, ACC);
    edge_scatter_k<<<gEW, 256, 0, stream>>>(srcp, dstp, HS, ACC);
    finalize_k<<<gNF, 256, 0, stream>>>(ACC, dinv, b1, H);

    // layer 2 (in-place on H is safe: finalize is elementwise from ACC)
    gemm_scale_k<<<gRow, 256, 0, stream>>>(H, W2, dinv, HS, ACC);
    edge_scatter_k<<<gEW, 256, 0, stream>>>(srcp, dstp, HS, ACC);
    finalize_k<<<gNF, 256, 0, stream>>>(ACC, dinv, b2, H);

    // global mean pool + classifier
    zero_pool_k<<<(N_GRAPHS * F + 255) / 256, 256, 0, stream>>>(pooled, counts);
    count_nodes_k<<<gN, 256, 0, stream>>>(bat, counts);
    pool_sum_k<<<gNF, 256, 0, stream>>>(bat, H, pooled);
    classify_k<<<1, 128, 0, stream>>>(pooled, counts, Wc, bc, out);
}